// CTCLayer_5385888989850
// MI455X (gfx1250) — compile-verified
//
#include <hip/hip_runtime.h>
#include <hip/hip_bf16.h>
#include <stdint.h>

#define PADV 99
#define NEGV (-1.0e9f)
#define LOG2E 1.4426950408889634f
#define LN2   0.6931471805599453f

#ifndef __has_builtin
#define __has_builtin(x) 0
#endif

// ---------- fast hardware transcendentals (v_exp_f32 / v_log_f32 are base-2) ----
__device__ __forceinline__ float fexp2(float x) {
#if __has_builtin(__builtin_amdgcn_exp2f)
  return __builtin_amdgcn_exp2f(x);
#else
  return exp2f(x);
#endif
}
__device__ __forceinline__ float flog2(float x) {
#if __has_builtin(__builtin_amdgcn_logf)
  return __builtin_amdgcn_logf(x);
#else
  return log2f(x);
#endif
}

// 3-way logaddexp in BASE-2 domain (inputs/outputs scaled by log2e).
// NEGV acts as -inf (exp2 underflows to 0). No muls on the dependent chain.
__device__ __forceinline__ float logaddexp3_2(float a, float b, float c) {
  float m = fmaxf(fmaxf(a, b), c);               // -> v_max3_num_f32
  float s = fexp2(a - m) + fexp2(b - m) + fexp2(c - m);
  return m + flog2(s);
}

// Unclamped lane gather: idx4 = srcLane*4 (precomputed, loop-invariant).
__device__ __forceinline__ float bperm(int idx4, float v) {
  return __int_as_float(__builtin_amdgcn_ds_bpermute(idx4, __float_as_int(v)));
}

// ---------- CDNA5 async global->LDS path (probe-verified signature) ----------
//   void __builtin_amdgcn_global_load_async_to_lds_b128(int4 AS(1)*, int4 AS(3)*, imm, imm)
#if __has_builtin(__builtin_amdgcn_global_load_async_to_lds_b128)
#define USE_ASYNC_LDS 1
typedef int v4i_t __attribute__((ext_vector_type(4)));
typedef __attribute__((address_space(1))) v4i_t GV4;
typedef __attribute__((address_space(3))) v4i_t LV4;
__device__ __forceinline__ void async_cp16(const float* g, float* l) {
  __builtin_amdgcn_global_load_async_to_lds_b128((GV4*)g, (LV4*)l, 0, 0);
}
#else
#define USE_ASYNC_LDS 0
#endif

#if __has_builtin(__builtin_amdgcn_s_wait_asynccnt)
#define WAIT_ASYNC2() __builtin_amdgcn_s_wait_asynccnt(2)
#else
#define WAIT_ASYNC2() asm volatile("s_wait_asynccnt 0x2" ::: "memory")
#endif

__device__ __forceinline__ void wait_ds0() {
#if __has_builtin(__builtin_amdgcn_s_wait_dscnt)
  __builtin_amdgcn_s_wait_dscnt(0);
#else
  asm volatile("s_wait_dscnt 0x0" ::: "memory");
#endif
}

// ============================================================================
// Kernel 1: lse2[b,t] = log2-sum-exp2 of y_pred[b,t,:]*log2e  (i.e. natural
// logsumexp scaled by LOG2E).  One wave per row of C=128.
// ============================================================================
__global__ __launch_bounds__(256) void ctc_lse_kernel(const float* __restrict__ yp,
                                                      float* __restrict__ lse_out,
                                                      int nrows) {
  int row  = blockIdx.x * 8 + (threadIdx.x >> 5);
  int lane = threadIdx.x & 31;
  if (row >= nrows) return;
  const float4 v = *(const float4*)(yp + (size_t)row * 128 + lane * 4);
  // work in base-2: x2 = x*LOG2E
  float ax = v.x * LOG2E, ay = v.y * LOG2E, az = v.z * LOG2E, aw = v.w * LOG2E;
  float m = fmaxf(fmaxf(ax, ay), fmaxf(az, aw));
#pragma unroll
  for (int off = 16; off; off >>= 1) m = fmaxf(m, __shfl_xor(m, off, 32));
  float s = fexp2(ax - m) + fexp2(ay - m) + fexp2(az - m) + fexp2(aw - m);
#pragma unroll
  for (int off = 16; off; off >>= 1) s += __shfl_xor(s, off, 32);
  if (lane == 0) lse_out[row] = m + flog2(s);   // base-2 lse
}

// ============================================================================
// Kernel 2: CTC forward scan (base-2 log domain). One wave32 per batch element.
// States s = 32*r + lane, r in {0,1,2} (s=64 lives in lane 0 of row 2).
// ============================================================================
__global__ __launch_bounds__(32) void ctc_scan_kernel(const int* __restrict__ yt,
                                                      const float* __restrict__ yp,
                                                      const float* __restrict__ lse,
                                                      float* __restrict__ loss) {
  const int b    = blockIdx.x;
  const int lane = threadIdx.x;
  const int T = 256, C = 128;

  // ---- labels & per-lane constants --------------------------------------
  int lab = yt[b * 32 + lane];
  unsigned long long bal = __ballot(lab != PADV);
  int llen = __popcll(bal);

  const bool odd = (lane & 1) != 0;
  int i0 = odd ? ((lane - 1) >> 1) : 0;   // label index for state s=lane (odd)
  int i1 = odd ? ((lane + 31) >> 1) : 0;  // label index for state s=lane+32 (odd)
  int lab0  = __shfl(lab, i0, 32);
  int lab1  = __shfl(lab, i1, 32);
  int lab0p = __shfl(lab, (odd && lane >= 3) ? (i0 - 1) : 0, 32);
  int lab1p = __shfl(lab, odd ? (i1 - 1) : 0, 32);
  int c0 = odd ? lab0 : (C - 1);          // class for state s=lane
  int c1 = odd ? lab1 : (C - 1);          // class for state s=lane+32
  bool skip0 = odd && ((lane == 1) ? true : (lab0 != lab0p));
  bool skip1 = odd && (lab1 != lab1p);

  const int lm1_4 = ((lane + 31) & 31) * 4;  // bpermute byte index, rotate-1
  const int lm2_4 = ((lane + 30) & 31) * 4;  // bpermute byte index, rotate-2

  const float* rowbase = yp + (size_t)b * T * C;
  const float* lrow    = lse + (size_t)b * T;

  float a0, a1, a2;   // alpha in base-2 log domain

#if USE_ASYNC_LDS
  __shared__ float rb[4][128];            // 2 KB ring of y_pred rows
  // prime rows 0..2 (prefetch distance 3)
#pragma unroll
  for (int i = 0; i < 3; ++i)
    async_cp16(rowbase + (size_t)i * C + lane * 4, &rb[i][lane * 4]);

  WAIT_ASYNC2();                           // row 0 ready
  {
    float lse0 = lrow[0];
    float vb = rb[0][C - 1];
    float v0 = rb[0][c0];
    float lpb = fmaf(vb, LOG2E, -lse0);
    float lp0 = fmaf(v0, LOG2E, -lse0);
    a0 = (lane == 0) ? lpb : ((lane == 1 && llen > 0) ? lp0 : NEGV);
    a1 = NEGV; a2 = NEGV;
  }
  wait_ds0();
  async_cp16(rowbase + (size_t)3 * C + lane * 4, &rb[3][lane * 4]);

  float lse_n = lrow[1];
#pragma unroll 4
  for (int t = 1; t < T; ++t) {
    WAIT_ASYNC2();                         // row t ready in slot t&3
    const float* rbuf = rb[t & 3];
    float lseT = lse_n;
    if (t + 1 < T) lse_n = lrow[t + 1];
    // lp in base-2 domain (off the critical chain: depends only on loads)
    float lpb = fmaf(rbuf[C - 1], LOG2E, -lseT);
    float lp0 = fmaf(rbuf[c0],    LOG2E, -lseT);
    float lp1 = fmaf(rbuf[c1],    LOG2E, -lseT);

    float r1a0 = bperm(lm1_4, a0), r1a1 = bperm(lm1_4, a1), r1a2 = bperm(lm1_4, a2);
    float r2a0 = bperm(lm2_4, a0), r2a1 = bperm(lm2_4, a1);
    float s1_0 = (lane == 0) ? NEGV : r1a0;
    float s1_1 = (lane == 0) ? r1a0 : r1a1;
    float s1_2 = (lane == 0) ? r1a1 : r1a2;
    float s2_0 = (lane < 2) ? NEGV : r2a0;
    float s2_1 = (lane < 2) ? r2a0 : r2a1;

    float n0 = logaddexp3_2(a0, s1_0, skip0 ? s2_0 : NEGV) + lp0;
    float n1 = logaddexp3_2(a1, s1_1, skip1 ? s2_1 : NEGV) + lp1;
    float n2 = logaddexp3_2(a2, s1_2, NEGV) + lpb;
    a0 = n0; a1 = n1; a2 = (lane == 0) ? n2 : NEGV;

    wait_ds0();                            // slot reads done before overwrite
    if (t + 3 < T)
      async_cp16(rowbase + (size_t)(t + 3) * C + lane * 4, &rb[(t + 3) & 3][lane * 4]);
  }
#else
  // Fallback: direct (L2-hot) global gathers with 1-step register prefetch.
  {
    float lse0 = lrow[0];
    float lpb = fmaf(rowbase[C - 1], LOG2E, -lse0);
    float lp0 = fmaf(rowbase[c0],    LOG2E, -lse0);
    a0 = (lane == 0) ? lpb : ((lane == 1 && llen > 0) ? lp0 : NEGV);
    a1 = NEGV; a2 = NEGV;
  }
  float cvb, cv0, cv1, clse;
  { const float* rp = rowbase + C; cvb = rp[C - 1]; cv0 = rp[c0]; cv1 = rp[c1]; clse = lrow[1]; }
  for (int t = 1; t < T; ++t) {
    float vb = cvb, v0 = cv0, v1 = cv1, lseT = clse;
    if (t + 1 < T) {
      const float* rp = rowbase + (size_t)(t + 1) * C;
      cvb = rp[C - 1]; cv0 = rp[c0]; cv1 = rp[c1]; clse = lrow[t + 1];
    }
    float lpb = fmaf(vb, LOG2E, -lseT);
    float lp0 = fmaf(v0, LOG2E, -lseT);
    float lp1 = fmaf(v1, LOG2E, -lseT);
    float r1a0 = bperm(lm1_4, a0), r1a1 = bperm(lm1_4, a1), r1a2 = bperm(lm1_4, a2);
    float r2a0 = bperm(lm2_4, a0), r2a1 = bperm(lm2_4, a1);
    float s1_0 = (lane == 0) ? NEGV : r1a0;
    float s1_1 = (lane == 0) ? r1a0 : r1a1;
    float s1_2 = (lane == 0) ? r1a1 : r1a2;
    float s2_0 = (lane < 2) ? NEGV : r2a0;
    float s2_1 = (lane < 2) ? r2a0 : r2a1;
    float n0 = logaddexp3_2(a0, s1_0, skip0 ? s2_0 : NEGV) + lp0;
    float n1 = logaddexp3_2(a1, s1_1, skip1 ? s2_1 : NEGV) + lp1;
    float n2 = logaddexp3_2(a2, s1_2, NEGV) + lpb;
    a0 = n0; a1 = n1; a2 = (lane == 0) ? n2 : NEGV;
  }
#endif

  // ---- final: -logaddexp(alpha[2L], alpha[2L-1]) in nats ------------------
  int idx = 2 * llen;                      // trailing blank state (<= 64)
  int r  = idx >> 5,  l  = idx & 31;
  float va  = (r == 0) ? a0 : ((r == 1) ? a1 : a2);
  float abl = __shfl(va, l, 32);
  int idx1 = (idx > 0) ? (idx - 1) : 0;
  int r1 = idx1 >> 5, l1 = idx1 & 31;
  float vc   = (r1 == 0) ? a0 : ((r1 == 1) ? a1 : a2);
  float albl = __shfl(vc, l1, 32);
  if (llen == 0) albl = NEGV;
  float lossb = -logaddexp3_2(abl, albl, NEGV) * LN2;   // back to natural log
  if (lane == 0) loss[b] = lossb;
}

// ============================================================================
// Kernel 3: deterministic mean over B losses.
// ============================================================================
__global__ __launch_bounds__(256) void ctc_mean_kernel(const float* __restrict__ loss,
                                                       float* __restrict__ out, int B) {
  __shared__ float sm[256];
  int t = threadIdx.x;
  float s = 0.0f;
  for (int i = t; i < B; i += 256) s += loss[i];
  sm[t] = s;
  __syncthreads();
#pragma unroll
  for (int off = 128; off; off >>= 1) {
    if (t < off) sm[t] += sm[t + off];
    __syncthreads();
  }
  if (t == 0) out[0] = sm[0] / (float)B;
}

extern "C" void kernel_launch(void* const* d_in, const int* in_sizes, int n_in,
                              void* d_out, int out_size, void* d_ws, size_t ws_size,
                              hipStream_t stream) {
  const int*   yt = (const int*)d_in[0];     // [B,32] int32
  const float* yp = (const float*)d_in[1];   // [B,256,128] float32
  const int B = in_sizes[0] / 32;
  const int T = 256;

  float* lse_ws  = (float*)d_ws;             // B*T floats (base-2 lse)
  float* loss_ws = lse_ws + (size_t)B * T;   // B floats

  int nrows = B * T;
  int blocks1 = (nrows + 7) / 8;
  ctc_lse_kernel<<<blocks1, 256, 0, stream>>>(yp, lse_ws, nrows);
  ctc_scan_kernel<<<B, 32, 0, stream>>>(yt, yp, lse_ws, loss_ws);
  ctc_mean_kernel<<<1, 256, 0, stream>>>(loss_ws, (float*)d_out, B);
}